// QKernel_3281355014837
// MI455X (gfx1250) — compile-verified
//
#include <hip/hip_runtime.h>

typedef __attribute__((ext_vector_type(2))) float v2f;
typedef __attribute__((ext_vector_type(8))) float v8f;

// ---------------------------------------------------------------------------
// Prelude: build the 16x16 complex circuit matrix M = (U0 (x) U1 (x) U2 (x) U3) * P_cnot
// from the 4x3 weights, and store Re/Im parts pre-swizzled into the WMMA
// A-operand layout for V_WMMA_F32_16X16X4_F32:
//   chunk c covers K = 4c..4c+3 ;  lane = half*16 + m (m = D row)
//   VGPR0 (float2.x) = A[m][4c + half*2], VGPR1 (float2.y) = A[m][4c + half*2 + 1]
// Additionally, M rows are permuted with rho (4..7 <-> 8..11) so that the
// (k, k+8) psi pairs needed by the Pauli reductions land in the same lane:
//   D rows 0..7  (lanes 0-15):  psi_0..3  in regs 0-3, psi_8..11  in regs 4-7
//   D rows 8..15 (lanes 16-31): psi_4..7  in regs 0-3, psi_12..15 in regs 4-7
// ws[0..255]   = packed Mr   (4 chunks * 32 lanes * float2)
// ws[256..511] = packed Mi
// ---------------------------------------------------------------------------
__global__ __launch_bounds__(256) void build_gate_matrix(const float* __restrict__ w,
                                                         float* __restrict__ ws)
{
    const int t = threadIdx.x;   // 0..255
    const int i = t >> 4;        // output row of M (psi index)
    const int j = t & 15;        // pre-permutation column

    // U3(theta,phi,omega) = [[c, -e^{i om} s], [e^{i ph} s, e^{i(ph+om)} c]]
    float ur[4][2][2], ui[4][2][2];
#pragma unroll
    for (int q = 0; q < 4; ++q) {
        float th = w[q * 3 + 0], ph = w[q * 3 + 1], om = w[q * 3 + 2];
        float c = cosf(th * 0.5f), s = sinf(th * 0.5f);
        float sp, cp, so, co, spo, cpo;
        sincosf(ph, &sp, &cp);
        sincosf(om, &so, &co);
        sincosf(ph + om, &spo, &cpo);
        ur[q][0][0] = c;        ui[q][0][0] = 0.0f;
        ur[q][0][1] = -co * s;  ui[q][0][1] = -so * s;
        ur[q][1][0] = cp * s;   ui[q][1][0] = sp * s;
        ur[q][1][1] = cpo * c;  ui[q][1][1] = spo * c;
    }

    // G[i][j] = prod_q U[q][bit_q(i)][bit_q(j)]   (qubit q <-> bit position 3-q)
    float gr = 1.0f, gi = 0.0f;
#pragma unroll
    for (int q = 0; q < 4; ++q) {
        int bi = (i >> (3 - q)) & 1;
        int bj = (j >> (3 - q)) & 1;
        float ar = ur[q][bi][bj], ai = ui[q][bi][bj];
        float nr = gr * ar - gi * ai;
        float ni = gr * ai + gi * ar;
        gr = nr; gi = ni;
    }

    // CNOT chain permutation: sigma(j) = f0(f1(f2(f3(j)))),
    // f_w: flip target bit (qubit (w+1)%4) if control bit (qubit w) is set.
    int k = j;
#pragma unroll
    for (int q = 3; q >= 0; --q) {
        int cb = 3 - q;
        int tb = 3 - ((q + 1) & 3);
        if ((k >> cb) & 1) k ^= (1 << tb);
    }

    // Row permutation rho: psi row i -> D row d (swap 4..7 <-> 8..11, involution)
    int d = i;
    if ((i >> 2) == 1) d = i + 4;
    else if ((i >> 2) == 2) d = i - 4;

    // M[i][sigma(j)] = G[i][j]  -> pack into A-operand layout at D row d
    const int c4 = k >> 2, half = (k >> 1) & 1, v = k & 1;
    const int idx = c4 * 64 + (half * 16 + d) * 2 + v;
    ws[idx]       = gr;
    ws[256 + idx] = gi;
}

// ---------------------------------------------------------------------------
// Main kernel: one wave32 per tile of 16 batch rows.
//   U(16x16) = Mr_perm * P_tile   via 4x v_wmma_f32_16x16x4_f32
//   V(16x16) = Mi_perm * P_tile   via 4x v_wmma_f32_16x16x4_f32
// B operand layout (4x16 f32): VGPR0 = K=4c+{0|2} (lanes 0-15|16-31), VGPR1 = +1.
// With the rho row permutation, each lane holds 4 (k, k+8) psi pairs in
// (reg r, reg r+4), so the Pauli reductions are lane-local except for one
// cross-half add per quantity (4 shuffles/tile total).
// Normalization folded in: M unitary => sum|q|^2 = |p|^2.
// ---------------------------------------------------------------------------
__global__ __launch_bounds__(256) void qexp_kernel(const float* __restrict__ patch,
                                                   const float* __restrict__ ws,
                                                   float* __restrict__ out,
                                                   int ntiles)
{
    const int lane = threadIdx.x & 31;
    const int wavesPerBlock = blockDim.x >> 5;
    const int wave = blockIdx.x * wavesPerBlock + (threadIdx.x >> 5);
    const int totalWaves = gridDim.x * wavesPerBlock;

    const int col  = lane & 15;                 // batch column within tile (N index)
    const int koff = (lane >= 16) ? 2 : 0;      // upper half-wave carries K+2,K+3

    // A operands: uniform for the whole dispatch, loaded once per wave.
    v2f ar[4], ai[4];
    const v2f* apr = (const v2f*)ws;
    const v2f* api = (const v2f*)(ws + 256);
#pragma unroll
    for (int c = 0; c < 4; ++c) {
        ar[c] = apr[c * 32 + lane];
        ai[c] = api[c * 32 + lane];
    }

    for (int tile = wave; tile < ntiles; tile += totalWaves) {
        const float* base = patch + (size_t)tile * 256;   // 16 rows * 16 floats

        v2f b[4];
#pragma unroll
        for (int c = 0; c < 4; ++c)
            b[c] = __builtin_nontemporal_load(
                       (const v2f*)(base + col * 16 + c * 4 + koff));

        v8f u = {};
        v8f vv = {};
#pragma unroll
        for (int c = 0; c < 4; ++c) {
            u  = __builtin_amdgcn_wmma_f32_16x16x4_f32(false, ar[c], false, b[c],
                                                       (short)0, u,  false, false);
            vv = __builtin_amdgcn_wmma_f32_16x16x4_f32(false, ai[c], false, b[c],
                                                       (short)0, vv, false, false);
        }

        // Lane-local pair reductions: regs (r, r+4) hold (psi_k, psi_{k+8}).
        float px = 0.0f, py = 0.0f, pa = 0.0f, pb = 0.0f;
#pragma unroll
        for (int r = 0; r < 4; ++r) {
            float ua = u[r],     va = vv[r];       // psi_k      (k < 8)
            float ub = u[r + 4], vb = vv[r + 4];   // psi_{k+8}
            px += ua * ub + va * vb;               // Re(conj(a) b)
            py += ua * vb - va * ub;               // Im(conj(a) b)
            pa += ua * ua + va * va;               // |a|^2
            pb += ub * ub + vb * vb;               // |b|^2
        }
        // Combine the two half-waves (each covers 4 of the 8 pairs).
        float cx = px + __shfl_xor(px, 16, 32);
        float cy = py + __shfl_xor(py, 16, 32);
        float ha = pa + __shfl_xor(pa, 16, 32);
        float hb = pb + __shfl_xor(pb, 16, 32);

        float invs = __builtin_amdgcn_rcpf(ha + hb);
        float ex = 2.0f * cx * invs;
        float ey = 2.0f * cy * invs;
        float ez = (ha - hb) * invs;

        if (lane < 16) {                          // both halves agree; write once
            float* o = out + ((size_t)tile * 16 + col) * 3;
            __builtin_nontemporal_store(ex, o + 0);
            __builtin_nontemporal_store(ey, o + 1);
            __builtin_nontemporal_store(ez, o + 2);
        }
    }
}

extern "C" void kernel_launch(void* const* d_in, const int* in_sizes, int n_in,
                              void* d_out, int out_size, void* d_ws, size_t ws_size,
                              hipStream_t stream) {
    const float* patch   = (const float*)d_in[0];   // (B, 16) float32
    const float* weights = (const float*)d_in[1];   // (4, 3)  float32
    float* out = (float*)d_out;                     // (B, 3)  float32
    float* ws  = (float*)d_ws;                      // needs 512 floats

    const int ntiles = in_sizes[0] / 256;           // B / 16

    build_gate_matrix<<<1, 256, 0, stream>>>(weights, ws);

    int blocks = 2048;                              // 16384 waves, grid-stride
    if (blocks * 8 > ntiles) blocks = (ntiles + 7) / 8;
    if (blocks < 1) blocks = 1;
    qexp_kernel<<<blocks, 256, 0, stream>>>(patch, ws, out, ntiles);
}